// HG_LGCN_vanilla_1279900254484
// MI455X (gfx1250) — compile-verified
//
#include <hip/hip_runtime.h>
#include <hip/hip_bf16.h>
#include <stdint.h>

// LightGCN forward on gfx1250 (MI455X):
//   out[:, 0:64]          = x
//   out[:, 64*(l+1)+0:64] = segment_sum(w_e * out[src_e, 64*l ...], dst_e)
//
// Roofline: all feature state (x + 4 slices = 102 MB) fits in the 192 MB L2.
// FLOPs are trivial (384 MFLOP). The push/atomic formulation costs 192M f32
// L2 atomics (~30x the cost of the actual data movement), so we build CSR by
// dst once per call (~2M int atomics) and run pull-based gather layers with
// ZERO atomics: 16 lanes per node, float4 accumulators in registers, one
// coalesced 256B row gather per incident edge, one float4 store per lane.
// Edge streams are staged through LDS with CDNA5 async-to-LDS (ASYNCcnt).
// WMMA is inapplicable: adjacency tile density ~1e-4.

#define D_FEAT   64
#define OUT_COLS 256   // (NUM_LAYERS+1)*D_FEAT
#define EPB      256   // edges per block (build kernels)
#define TPB      256
#define SCAN_T   1024

// ---------------------------------------------------------------------------
// Init: copy x into cols [0,64); optionally zero cols [64,256) (atomic path).
// COLS4 = number of float4 columns written per row (16 = copy only, 64 = all).
// ---------------------------------------------------------------------------
template <int COLS4>
__global__ void lgcn_init(const float* __restrict__ x,
                          float* __restrict__ out,
                          int n_nodes) {
  size_t tid   = (size_t)blockIdx.x * TPB + threadIdx.x;
  size_t total = (size_t)n_nodes * COLS4;
  if (tid >= total) return;
  size_t node = tid / COLS4;
  int    c    = (int)(tid % COLS4);
  float4 v;
  if (c < D_FEAT / 4) {
    v = ((const float4*)(x + node * D_FEAT))[c];
  } else {
    v = make_float4(0.f, 0.f, 0.f, 0.f);
  }
  ((float4*)(out + node * OUT_COLS))[c] = v;
}

// ---------------------------------------------------------------------------
// CSR build step 1: in-degree histogram. dst tile staged via async-to-LDS.
// ---------------------------------------------------------------------------
__global__ void degree_count(const int* __restrict__ dst,
                             int* __restrict__ deg, int n_edges) {
  __shared__ int s_dst[EPB];
  const int t = threadIdx.x;
  const int idx = blockIdx.x * EPB + t;
  const int cidx = (idx < n_edges) ? idx : (n_edges - 1);

  uint32_t lds = (uint32_t)(uintptr_t)&s_dst[t];
  const int* g = dst + cidx;
  asm volatile("global_load_async_to_lds_b32 %0, %1, off"
               :: "v"(lds), "v"(g) : "memory");
  asm volatile("s_wait_asynccnt 0" ::: "memory");
  __syncthreads();

  if (idx < n_edges) atomicAdd(&deg[s_dst[t]], 1);
}

// ---------------------------------------------------------------------------
// CSR build step 2: exclusive scan of deg[0..n) -> rowptr[0..n]. Single block:
// each thread serially sums a contiguous chunk, Hillis-Steele block scan of
// the 1024 chunk sums in LDS, then serial exclusive write-out.
// ---------------------------------------------------------------------------
__global__ void scan_exclusive(const int* __restrict__ deg,
                               int* __restrict__ rowptr, int n) {
  __shared__ int sums[SCAN_T];
  const int t = threadIdx.x;
  const int chunk = (n + SCAN_T - 1) / SCAN_T;
  int begin = t * chunk;
  int end   = begin + chunk; if (end > n) end = n;

  int s = 0;
  for (int i = begin; i < end; ++i) s += deg[i];
  sums[t] = s;
  __syncthreads();

  for (int off = 1; off < SCAN_T; off <<= 1) {   // inclusive scan
    int v = (t >= off) ? sums[t - off] : 0;
    __syncthreads();
    sums[t] += v;
    __syncthreads();
  }

  int prefix = (t == 0) ? 0 : sums[t - 1];
  for (int i = begin; i < end; ++i) {
    rowptr[i] = prefix;
    prefix += deg[i];
  }
  if (t == SCAN_T - 1) rowptr[n] = prefix;       // total edge count
}

// ---------------------------------------------------------------------------
// CSR build step 3: compaction fill. (src,dst,w) tiles staged via async-to-LDS,
// then each thread places its edge at rowptr[dst] + cursor[dst]++.
// ---------------------------------------------------------------------------
__global__ void csr_fill(const int* __restrict__ src,
                         const int* __restrict__ dst,
                         const float* __restrict__ w,
                         const int* __restrict__ rowptr,
                         int* __restrict__ cursor,
                         int* __restrict__ csr_src,
                         float* __restrict__ csr_w, int n_edges) {
  __shared__ int   s_src[EPB];
  __shared__ int   s_dst[EPB];
  __shared__ float s_w[EPB];
  const int t = threadIdx.x;
  const int idx = blockIdx.x * EPB + t;
  const int cidx = (idx < n_edges) ? idx : (n_edges - 1);

  uint32_t lds_s = (uint32_t)(uintptr_t)&s_src[t];
  uint32_t lds_d = (uint32_t)(uintptr_t)&s_dst[t];
  uint32_t lds_w = (uint32_t)(uintptr_t)&s_w[t];
  const int*   gs = src + cidx;
  const int*   gd = dst + cidx;
  const float* gw = w + cidx;
  asm volatile("global_load_async_to_lds_b32 %0, %1, off"
               :: "v"(lds_s), "v"(gs) : "memory");
  asm volatile("global_load_async_to_lds_b32 %0, %1, off"
               :: "v"(lds_d), "v"(gd) : "memory");
  asm volatile("global_load_async_to_lds_b32 %0, %1, off"
               :: "v"(lds_w), "v"(gw) : "memory");
  asm volatile("s_wait_asynccnt 0" ::: "memory");
  __syncthreads();

  if (idx < n_edges) {
    int d   = s_dst[t];
    int pos = rowptr[d] + atomicAdd(&cursor[d], 1);
    csr_src[pos] = s_src[t];
    csr_w[pos]   = s_w[t];
  }
}

// ---------------------------------------------------------------------------
// Pull-based layer: 16 lanes per node; lane owns float4 chunk of the 64-wide
// row. Serial loop over incident edges: broadcast csr_src/csr_w (same address
// across the 16-lane group), coalesced 256B gather of the source row, FMA into
// registers. One float4 store per lane; no atomics, no read-modify-write.
// Nodes with zero incident edges store zeros (matches segment_sum).
// ---------------------------------------------------------------------------
__global__ void lgcn_gather(const int* __restrict__ rowptr,
                            const int* __restrict__ csr_src,
                            const float* __restrict__ csr_w,
                            float* __restrict__ out,
                            int n_nodes, int layer) {
  int gt   = blockIdx.x * TPB + threadIdx.x;
  int node = gt >> 4;
  int lane = gt & 15;
  if (node >= n_nodes) return;

  const int beg = rowptr[node];
  const int end = rowptr[node + 1];
  const int in_off = layer * D_FEAT;

  float4 acc = make_float4(0.f, 0.f, 0.f, 0.f);
  for (int e = beg; e < end; ++e) {
    int   s  = csr_src[e];
    float ww = csr_w[e];
    float4 v = ((const float4*)(out + (size_t)s * OUT_COLS + in_off))[lane];
    acc.x = fmaf(ww, v.x, acc.x);
    acc.y = fmaf(ww, v.y, acc.y);
    acc.z = fmaf(ww, v.z, acc.z);
    acc.w = fmaf(ww, v.w, acc.w);
  }
  ((float4*)(out + (size_t)node * OUT_COLS + in_off + D_FEAT))[lane] = acc;
}

// ---------------------------------------------------------------------------
// Fallback push path (used only if ws_size is too small for CSR): per-edge
// scatter with f32 atomics; edge tile staged via async-to-LDS.
// ---------------------------------------------------------------------------
__global__ void lgcn_scatter(const int* __restrict__ src,
                             const int* __restrict__ dst,
                             const float* __restrict__ w,
                             float* __restrict__ out,
                             int n_edges, int layer) {
  __shared__ int   s_src[EPB];
  __shared__ int   s_dst[EPB];
  __shared__ float s_w[EPB];
  const int t  = threadIdx.x;
  const int e0 = blockIdx.x * EPB;
  int idx  = e0 + t;
  int cidx = (idx < n_edges) ? idx : (n_edges - 1);

  uint32_t lds_s = (uint32_t)(uintptr_t)&s_src[t];
  uint32_t lds_d = (uint32_t)(uintptr_t)&s_dst[t];
  uint32_t lds_w = (uint32_t)(uintptr_t)&s_w[t];
  const int*   gs = src + cidx;
  const int*   gd = dst + cidx;
  const float* gw = w + cidx;
  asm volatile("global_load_async_to_lds_b32 %0, %1, off"
               :: "v"(lds_s), "v"(gs) : "memory");
  asm volatile("global_load_async_to_lds_b32 %0, %1, off"
               :: "v"(lds_d), "v"(gd) : "memory");
  asm volatile("global_load_async_to_lds_b32 %0, %1, off"
               :: "v"(lds_w), "v"(gw) : "memory");
  asm volatile("s_wait_asynccnt 0" ::: "memory");
  __syncthreads();

  const int in_off  = layer * D_FEAT;
  const int out_off = in_off + D_FEAT;
  #pragma unroll 4
  for (int it = 0; it < (EPB * 16) / TPB; ++it) {
    int work = it * TPB + t;
    int el   = work >> 4;
    int ch   = work & 15;
    if (e0 + el < n_edges) {
      int   s  = s_src[el];
      int   d  = s_dst[el];
      float ww = s_w[el];
      float4 v = ((const float4*)(out + (size_t)s * OUT_COLS + in_off))[ch];
      float* o = out + (size_t)d * OUT_COLS + out_off + ch * 4;
      atomicAdd(o + 0, ww * v.x);
      atomicAdd(o + 1, ww * v.y);
      atomicAdd(o + 2, ww * v.z);
      atomicAdd(o + 3, ww * v.w);
    }
  }
}

// ---------------------------------------------------------------------------
// Launch. CSR path (preferred): memset counters -> degree -> scan -> fill ->
// copy x -> 3 gather layers. All ws state rewritten every call (replay-safe).
// ---------------------------------------------------------------------------
extern "C" void kernel_launch(void* const* d_in, const int* in_sizes, int n_in,
                              void* d_out, int out_size, void* d_ws, size_t ws_size,
                              hipStream_t stream) {
  const float* x  = (const float*)d_in[0];   // [N, 64]
  const int*   ei = (const int*)d_in[1];     // [2, E] flat: row0=src, row1=dst
  const float* ew = (const float*)d_in[2];   // [E]
  // d_in[3] (edge_type) unused: sum over both partitions == full aggregation.

  const int E = in_sizes[2];
  const int N = in_sizes[0] / D_FEAT;
  const int* srcv = ei;
  const int* dstv = ei + E;
  float* out = (float*)d_out;

  const int edge_blocks = (E + EPB - 1) / EPB;

  // Workspace layout (ints): deg[N] | cursor[N] | rowptr[N+1] | csr_src[E] | csr_w[E]
  const size_t need = ((size_t)3 * N + 1 + 2 * (size_t)E) * sizeof(int);

  if (ws_size >= need) {
    int*   deg     = (int*)d_ws;
    int*   cursor  = deg + N;
    int*   rowptr  = cursor + N;
    int*   csr_src = rowptr + (N + 1);
    float* csr_w   = (float*)(csr_src + E);

    hipMemsetAsync(deg, 0, (size_t)2 * N * sizeof(int), stream); // deg + cursor
    degree_count<<<edge_blocks, TPB, 0, stream>>>(dstv, deg, E);
    scan_exclusive<<<1, SCAN_T, 0, stream>>>(deg, rowptr, N);
    csr_fill<<<edge_blocks, TPB, 0, stream>>>(srcv, dstv, ew, rowptr, cursor,
                                              csr_src, csr_w, E);

    size_t init_items = (size_t)N * 16;   // copy x only; gather writes the rest
    lgcn_init<16><<<(int)((init_items + TPB - 1) / TPB), TPB, 0, stream>>>(x, out, N);

    int gather_blocks = (int)(((size_t)N * 16 + TPB - 1) / TPB);
    for (int l = 0; l < 3; ++l) {
      lgcn_gather<<<gather_blocks, TPB, 0, stream>>>(rowptr, csr_src, csr_w,
                                                     out, N, l);
    }
  } else {
    // Fallback: atomic push path, no workspace required.
    size_t init_items = (size_t)N * 64;   // copy x + zero accumulation slices
    lgcn_init<64><<<(int)((init_items + TPB - 1) / TPB), TPB, 0, stream>>>(x, out, N);
    for (int l = 0; l < 3; ++l) {
      lgcn_scatter<<<edge_blocks, TPB, 0, stream>>>(srcv, dstv, ew, out, E, l);
    }
  }
}